// MyModel_51797305589802
// MI455X (gfx1250) — compile-verified
//
#include <hip/hip_runtime.h>

typedef __attribute__((ext_vector_type(2))) float v2f;
typedef __attribute__((ext_vector_type(8))) float v8f;

#define NV 64
#define WPB 8   // waves per block (256 threads)

// One wave32 processes one batch:
//   adj_hat = A + I; dinv = rowsum(adj_hat)^-1/2
//   S = (dinv .* adj_hat .* dinv^T) @ (h0 @ W1)   via V_WMMA_F32_16X16X4_F32
//   out[b] = (sum_v relu(S))[0:3] @ W2
__global__ void __launch_bounds__(256)
gcn_pool_kernel(const float* __restrict__ A, const float* __restrict__ h0,
                const float* __restrict__ W1, const float* __restrict__ W2,
                float* __restrict__ out, int B)
{
  __shared__ float dinvS[WPB][NV];
  __shared__ float GS[WPB][NV * 4];   // G[w][f], f in 0..2, slot 3 unused

  const int tid  = threadIdx.x;
  const int wv   = tid >> 5;
  const int lane = tid & 31;
  const int r    = lane & 15;   // M row within tile / N column for B,C
  const int h    = lane >> 4;   // lane half -> K pairing

  const int b = blockIdx.x * WPB + wv;
  if (b >= B) return;           // wave-uniform: EXEC stays all-ones below

  const float* Ab = A  + (size_t)b * (NV * NV);
  const float* Hb = h0 + (size_t)b * (NV * 5);

  // ---- Phase 1: A fragments in WMMA layout, identity fused -----------------
  // a[m][k]: 16x4 tile, rows 16m+r, cols {4k+2h, 4k+2h+1}
  v2f a[4][16];
#pragma unroll
  for (int m = 0; m < 4; ++m) {
    const int row = 16 * m + r;
#pragma unroll
    for (int k = 0; k < 16; ++k) {
      const int col = 4 * k + 2 * h;
      float x = Ab[row * NV + col];
      float y = Ab[row * NV + col + 1];
      if (row == col)     x += 1.0f;
      if (row == col + 1) y += 1.0f;
      a[m][k].x = x;
      a[m][k].y = y;
    }
  }

  // ---- Phase 2: rowsums -> dinv; fold dinv[row] into A fragments ----------
  float dv[4];
#pragma unroll
  for (int m = 0; m < 4; ++m) {
    float p = 0.0f;
#pragma unroll
    for (int k = 0; k < 16; ++k) p += a[m][k].x + a[m][k].y;
    p += __shfl_xor(p, 16, 32);        // combine column halves
    dv[m] = 1.0f / sqrtf(p);           // rowsum^-1/2 for row 16m + r
  }
  if (h == 0) {                        // publish all 64 dinv values
#pragma unroll
    for (int m = 0; m < 4; ++m) dinvS[wv][16 * m + r] = dv[m];
  }
#pragma unroll
  for (int m = 0; m < 4; ++m)
#pragma unroll
    for (int k = 0; k < 16; ++k) { a[m][k].x *= dv[m]; a[m][k].y *= dv[m]; }

  // ---- Phase 3: G = dinv .* (h0 @ W1), rows w = lane and lane+32 ----------
  float w1[5][3];
#pragma unroll
  for (int c = 0; c < 5; ++c)
#pragma unroll
    for (int f = 0; f < 3; ++f) w1[c][f] = W1[c * 3 + f];

#pragma unroll
  for (int t = 0; t < 2; ++t) {
    const int w = lane + 32 * t;
    const float* hr = Hb + w * 5;
    float hv[5];
#pragma unroll
    for (int c = 0; c < 5; ++c) hv[c] = hr[c];
    const float dw = dinvS[wv][w];
#pragma unroll
    for (int f = 0; f < 3; ++f) {
      float s = 0.0f;
#pragma unroll
      for (int c = 0; c < 5; ++c) s += hv[c] * w1[c][f];
      GS[wv][w * 4 + f] = dw * s;
    }
  }

  // ---- Phase 4: B fragments (4x16, K = 4k+2h+j, N = r; N>=3 padded 0) -----
  v2f bf[16];
#pragma unroll
  for (int k = 0; k < 16; ++k) {
    const int w0 = 4 * k + 2 * h;
    float bx = 0.0f, by = 0.0f;
    if (r < 3) {
      bx = GS[wv][w0 * 4 + r];
      by = GS[wv][(w0 + 1) * 4 + r];
    }
    bf[k].x = bx;
    bf[k].y = by;
  }

  // ---- Phase 5: S = A_nrm @ G via WMMA; relu + node pooling ---------------
  float p = 0.0f;
#pragma unroll
  for (int m = 0; m < 4; ++m) {
    v8f acc = {0.f, 0.f, 0.f, 0.f, 0.f, 0.f, 0.f, 0.f};
#pragma unroll
    for (int k = 0; k < 16; ++k)
      acc = __builtin_amdgcn_wmma_f32_16x16x4_f32(
          /*neg_a=*/false, a[m][k], /*neg_b=*/false, bf[k],
          /*c_mod=*/(short)0, acc, /*reuse_a=*/false, /*reuse_b=*/false);
    // acc[i] = S[16m + i + 8h][r]
#pragma unroll
    for (int i = 0; i < 8; ++i) p += fmaxf(acc[i], 0.0f);
  }
  p += __shfl_xor(p, 16, 32);          // pooled[r] now in lanes r and r+16

  // ---- Phase 6: out[b] = pooled @ W2 --------------------------------------
  float t = 0.0f;
  if (lane < 3) t = p * W2[lane];
  float o = __shfl(t, 0, 32) + __shfl(t, 1, 32) + __shfl(t, 2, 32);
  if (lane == 0) out[b] = o;
}

extern "C" void kernel_launch(void* const* d_in, const int* in_sizes, int n_in,
                              void* d_out, int out_size, void* d_ws, size_t ws_size,
                              hipStream_t stream) {
  const float* A  = (const float*)d_in[0];
  const float* h0 = (const float*)d_in[1];
  const float* W1 = (const float*)d_in[2];
  const float* W2 = (const float*)d_in[3];
  float* out = (float*)d_out;

  const int B = in_sizes[0] / (NV * NV);
  dim3 block(256);
  dim3 grid((B + WPB - 1) / WPB);
  gcn_pool_kernel<<<grid, block, 0, stream>>>(A, h0, W1, W2, out, B);
}